// GNN_4655744549280
// MI455X (gfx1250) — compile-verified
//
#include <hip/hip_runtime.h>

// ---------------------------------------------------------------------------
// 2-layer GCN, fused for MI455X (gfx1250, wave32).
//   h  = (x @ W) * dinv[row]                     (WMMA f32 16x16x4, exact fp32)
//   out[j] = relu?( dinv[j] * (sum_{i->j} h[i] + h[j]) + b )   (CSR aggregate)
// CSR built per call: degree histogram -> hierarchical scan -> cursor fill.
// W repacked once per call into B-fragment order (coalesced b64 loads).
// ---------------------------------------------------------------------------

typedef __attribute__((ext_vector_type(2))) float v2f;
typedef __attribute__((ext_vector_type(4))) float v4f;
typedef __attribute__((ext_vector_type(8))) float v8f;

template <int V> struct VecT;
template <> struct VecT<2> { typedef v2f type; };
template <> struct VecT<4> { typedef v4f type; };

// ---- degree histogram over destinations (self-loops added analytically) ----
__global__ void k_deg(const long long* __restrict__ dst, unsigned* __restrict__ deg, int E) {
  int e = blockIdx.x * blockDim.x + threadIdx.x;
  if (e < E) atomicAdd(&deg[(int)dst[e]], 1u);
}

// ---- dinv = rsqrt(deg + 1)  (+1 is the self-loop; always > 0) ----
__global__ void k_dinv(const unsigned* __restrict__ deg, float* __restrict__ dinv, int n) {
  int i = blockIdx.x * blockDim.x + threadIdx.x;
  if (i < n) dinv[i] = rsqrtf((float)(deg[i] + 1u));
}

// ---- 256-thread block exclusive scan from barrier-free wave32 shfl scans ----
__device__ inline unsigned block_exscan256(unsigned v, unsigned& blockTotal) {
  __shared__ unsigned wsum[8];
  __shared__ unsigned btot;
  unsigned s = v;                               // inclusive scan within wave32
  #pragma unroll
  for (int d = 1; d < 32; d <<= 1) {
    unsigned t = __shfl_up(s, d, 32);
    if (((int)threadIdx.x & 31) >= d) s += t;
  }
  int wid = (int)threadIdx.x >> 5;
  if (((int)threadIdx.x & 31) == 31) wsum[wid] = s;
  __syncthreads();
  if (threadIdx.x < 8) {                        // wave 0 scans the 8 wave sums
    unsigned w = wsum[threadIdx.x];
    unsigned ws2 = w;
    #pragma unroll
    for (int d = 1; d < 8; d <<= 1) {
      unsigned t = __shfl_up(ws2, d, 32);
      if ((int)threadIdx.x >= d) ws2 += t;
    }
    wsum[threadIdx.x] = ws2 - w;                // exclusive prefix of wave sums
    if (threadIdx.x == 7) btot = ws2;           // block total
  }
  __syncthreads();
  blockTotal = btot;
  return wsum[wid] + s - v;                     // exclusive prefix within block
}

// ---- scan pass 1: per-block exclusive scan + block sums ----
__global__ void k_scan1(const unsigned* __restrict__ deg, unsigned* __restrict__ offs,
                        unsigned* __restrict__ blockSums, int n) {
  int i = blockIdx.x * 256 + (int)threadIdx.x;
  unsigned v = (i < n) ? deg[i] : 0u;
  unsigned tot;
  unsigned excl = block_exscan256(v, tot);
  if (i < n) offs[i] = excl;
  if (threadIdx.x == 0) blockSums[blockIdx.x] = tot;
}

// ---- scan pass 2: single block scans the (<=256) block sums ----
__global__ void k_scan2(const unsigned* __restrict__ blockSums, unsigned* __restrict__ blockPrefix,
                        unsigned* __restrict__ totalOut, int nb) {
  unsigned v = ((int)threadIdx.x < nb) ? blockSums[threadIdx.x] : 0u;
  unsigned tot;
  unsigned excl = block_exscan256(v, tot);
  if ((int)threadIdx.x < nb) blockPrefix[threadIdx.x] = excl;
  if (threadIdx.x == 0) totalOut[0] = tot;
}

// ---- scan pass 3: uniform add of block prefix; append grand total ----
__global__ void k_scan3(unsigned* __restrict__ offs, const unsigned* __restrict__ blockPrefix,
                        const unsigned* __restrict__ totalOut, int n) {
  int i = blockIdx.x * 256 + (int)threadIdx.x;
  if (i < n) offs[i] += blockPrefix[blockIdx.x];
  if (i == 0) offs[n] = totalOut[0];
}

// ---- scatter src indices into CSR slots via per-node cursors ----
__global__ void k_fill(const long long* __restrict__ src, const long long* __restrict__ dst,
                       unsigned* __restrict__ cursor, int* __restrict__ csr, int E) {
  int e = blockIdx.x * blockDim.x + threadIdx.x;
  if (e >= E) return;
  unsigned p = atomicAdd(&cursor[(int)dst[e]], 1u);
  csr[p] = (int)src[e];
}

// ---- repack W[K,N] into B-fragment order:
//  Wp[((kg*(N/16)+t)*32 + l)*2 + p] = W[(4*kg + 2*(l>>4) + p)*N + t*16 + (l&15)]
// so each lane's B fragment for (k-group kg, 16-col tile t) is one aligned float2.
__global__ void k_packW(const float* __restrict__ W, float* __restrict__ Wp, int K, int N) {
  int idx = blockIdx.x * blockDim.x + threadIdx.x;
  if (idx >= K * N) return;
  int p    = idx & 1;
  int l    = (idx >> 1) & 31;
  int rest = idx >> 6;
  int tiles = N >> 4;
  int t  = rest % tiles;
  int kg = rest / tiles;
  int k = 4 * kg + ((l >> 4) << 1) + p;
  int c = t * 16 + (l & 15);
  Wp[idx] = W[(size_t)k * N + c];
}

// ---- GEMM: out[M,N] = (A[M,K] @ W[K,N]) * scale[row]; one wave => 16x64 tile
// Requires M % 16 == 0, K % 4 == 0, N % 64 == 0 (true here: 50000, 128, 128/64).
__global__ void k_gemm(const float* __restrict__ A, const float* __restrict__ Wp,
                       const float* __restrict__ scale, float* __restrict__ out,
                       int M, int K, int N) {
  int wave = blockIdx.x * (blockDim.x >> 5) + ((int)threadIdx.x >> 5);
  int lane = (int)threadIdx.x & 31;
  int groups = N >> 6;                       // 64-wide column groups
  int tiles  = N >> 4;                       // 16-wide column tiles
  int tm = wave / groups;
  int g  = wave % groups;
  if (tm * 16 >= M) return;                  // wave-uniform: EXEC stays all-ones

  int row  = tm * 16 + (lane & 15);
  int koff = (lane >> 4) << 1;               // lanes 16-31 hold K+2,K+3

  v8f c0{}, c1{}, c2{}, c3{};
  for (int k0 = 0, kg = 0; k0 < K; k0 += 4, ++kg) {
    v2f a = *(const v2f*)(A + (size_t)row * K + k0 + koff);          // b64
    const float* wp = Wp + (((size_t)kg * tiles + g * 4) << 6) + lane * 2;
    v2f b0 = *(const v2f*)(wp);                                      // b64, coalesced
    v2f b1 = *(const v2f*)(wp + 64);
    v2f b2 = *(const v2f*)(wp + 128);
    v2f b3 = *(const v2f*)(wp + 192);
    c0 = __builtin_amdgcn_wmma_f32_16x16x4_f32(false, a, false, b0, (short)0, c0, false, false);
    c1 = __builtin_amdgcn_wmma_f32_16x16x4_f32(false, a, false, b1, (short)0, c1, false, false);
    c2 = __builtin_amdgcn_wmma_f32_16x16x4_f32(false, a, false, b2, (short)0, c2, false, false);
    c3 = __builtin_amdgcn_wmma_f32_16x16x4_f32(false, a, false, b3, (short)0, c3, false, false);
  }

  int orow = tm * 16 + ((lane >> 4) << 3);   // VGPR r: lanes0-15 M=r, lanes16-31 M=8+r
  int ocol = g * 64 + (lane & 15);
  for (int r = 0; r < 8; ++r) {
    float s = scale[orow + r];
    size_t base = (size_t)(orow + r) * N + ocol;
    out[base]      = c0[r] * s;
    out[base + 16] = c1[r] * s;
    out[base + 32] = c2[r] * s;
    out[base + 48] = c3[r] * s;
  }
}

// ---- CSR aggregation: one wave per node; F features = F/32 floats per lane --
template <int F, bool RELU>
__global__ void k_agg(const float* __restrict__ hs, const unsigned* __restrict__ offs,
                      const int* __restrict__ csr, const float* __restrict__ dinv,
                      const float* __restrict__ bias, float* __restrict__ out, int n) {
  constexpr int V = F / 32;
  typedef typename VecT<V>::type vec;
  int j    = blockIdx.x * (blockDim.x >> 5) + ((int)threadIdx.x >> 5);
  int lane = (int)threadIdx.x & 31;
  if (j >= n) return;
  int c = lane * V;

  vec acc = *(const vec*)(hs + (size_t)j * F + c);         // self-loop term
  vec acc2{};
  unsigned e0 = offs[j], e1 = offs[j + 1];
  unsigned e = e0;
  for (; e + 2 <= e1; e += 2) {                            // 2 independent chains
    int s0 = csr[e], s1 = csr[e + 1];
    if (e + 4 <= e1) {
      __builtin_prefetch(hs + (size_t)csr[e + 2] * F + c, 0, 0);  // global_prefetch_b8
      __builtin_prefetch(hs + (size_t)csr[e + 3] * F + c, 0, 0);
    }
    acc  += *(const vec*)(hs + (size_t)s0 * F + c);
    acc2 += *(const vec*)(hs + (size_t)s1 * F + c);
  }
  if (e < e1) acc += *(const vec*)(hs + (size_t)csr[e] * F + c);
  acc += acc2;

  float dj = dinv[j];
  vec b = *(const vec*)(bias + c);
  vec r = acc * dj + b;
  if (RELU) {
    #pragma unroll
    for (int i = 0; i < V; ++i) r[i] = r[i] > 0.f ? r[i] : 0.f;
  }
  *(vec*)(out + (size_t)j * F + c) = r;
}

// ---------------------------------------------------------------------------
extern "C" void kernel_launch(void* const* d_in, const int* in_sizes, int n_in,
                              void* d_out, int out_size, void* d_ws, size_t ws_size,
                              hipStream_t stream) {
  const float*     x   = (const float*)d_in[0];
  const long long* ei  = (const long long*)d_in[1];      // int64 [2, E]
  const float*     W1  = (const float*)d_in[2];
  const float*     b1  = (const float*)d_in[3];
  const float*     W2  = (const float*)d_in[4];
  const float*     b2  = (const float*)d_in[5];
  float*           outp = (float*)d_out;

  const int d_hid  = in_sizes[3];                 // 128
  const int d_outf = in_sizes[5];                 // 64
  const int d_inf  = in_sizes[2] / d_hid;         // 128
  const int Nn     = in_sizes[0] / d_inf;         // 50000
  const int E      = in_sizes[1] / 2;             // 800000
  const long long* srcIdx = ei;
  const long long* dstIdx = ei + E;

  // --- carve workspace (256B-aligned slabs) ---
  char* ws = (char*)d_ws;
  auto take = [&](size_t bytes) -> void* {
    void* p = (void*)ws;
    ws += (bytes + 255) & ~(size_t)255;
    return p;
  };
  unsigned* deg      = (unsigned*)take((size_t)Nn * 4);
  unsigned* offs     = (unsigned*)take(((size_t)Nn + 1) * 4);
  unsigned* cursor   = (unsigned*)take((size_t)Nn * 4);
  float*    dinv     = (float*)take((size_t)Nn * 4);
  int*      csr      = (int*)take((size_t)E * 4);
  float*    hs1      = (float*)take((size_t)Nn * d_hid * 4);
  float*    out1     = (float*)take((size_t)Nn * d_hid * 4);
  float*    hs2      = (float*)take((size_t)Nn * d_outf * 4);
  float*    Wp1      = (float*)take((size_t)d_inf * d_hid * 4);
  float*    Wp2      = (float*)take((size_t)d_hid * d_outf * 4);
  unsigned* blockSums   = (unsigned*)take(256 * 4);
  unsigned* blockPrefix = (unsigned*)take(256 * 4);
  unsigned* totalOut    = (unsigned*)take(4);
  (void)ws_size; (void)n_in; (void)out_size;

  const int scanBlocks = (Nn + 255) / 256;        // 196 (<=256 required)

  // --- graph structure (recomputed every call; deterministic output) ---
  hipMemsetAsync(deg, 0, (size_t)Nn * 4, stream);
  k_deg <<<(E + 255) / 256, 256, 0, stream>>>(dstIdx, deg, E);
  k_dinv<<<(Nn + 255) / 256, 256, 0, stream>>>(deg, dinv, Nn);
  k_scan1<<<scanBlocks, 256, 0, stream>>>(deg, offs, blockSums, Nn);
  k_scan2<<<1, 256, 0, stream>>>(blockSums, blockPrefix, totalOut, scanBlocks);
  k_scan3<<<scanBlocks, 256, 0, stream>>>(offs, blockPrefix, totalOut, Nn);
  hipMemcpyAsync(cursor, offs, (size_t)Nn * 4, hipMemcpyDeviceToDevice, stream);
  k_fill<<<(E + 255) / 256, 256, 0, stream>>>(srcIdx, dstIdx, cursor, csr, E);

  // --- repack weights into B-fragment order (tiny; once per call) ---
  k_packW<<<(d_inf * d_hid + 255) / 256, 256, 0, stream>>>(W1, Wp1, d_inf, d_hid);
  k_packW<<<(d_hid * d_outf + 255) / 256, 256, 0, stream>>>(W2, Wp2, d_hid, d_outf);

  // --- layer 1: hs1 = (x @ W1) * dinv ; out1 = relu(dinv*(Σ hs1) + b1) ---
  {
    int waves = (Nn / 16) * (d_hid / 64);
    int blocks = (waves * 32 + 255) / 256;
    k_gemm<<<blocks, 256, 0, stream>>>(x, Wp1, dinv, hs1, Nn, d_inf, d_hid);
    int ablocks = (Nn + 7) / 8;                   // 8 waves of 32 per block
    k_agg<128, true><<<ablocks, 256, 0, stream>>>(hs1, offs, csr, dinv, b1, out1, Nn);
  }

  // --- layer 2: hs2 = (out1 @ W2) * dinv ; out = dinv*(Σ hs2) + b2 ---
  {
    int waves = (Nn / 16) * (d_outf / 64);
    int blocks = (waves * 32 + 255) / 256;
    k_gemm<<<blocks, 256, 0, stream>>>(out1, Wp2, dinv, hs2, Nn, d_hid, d_outf);
    int ablocks = (Nn + 7) / 8;
    k_agg<64, false><<<ablocks, 256, 0, stream>>>(hs2, offs, csr, dinv, b2, outp, Nn);
  }
}